// Model_12429635354725
// MI455X (gfx1250) — compile-verified
//
#include <hip/hip_runtime.h>
#include <hip/hip_bf16.h>

// GRU: I=8, H=16, T=512, B=8192, OUT=10.
// One wave32 handles a 16-batch tile; per timestep 18 V_WMMA_F32_16X16X4_F32
// (fp32 matrix cores) accumulate the gate pre-activations; h is re-laid out
// C-layout -> A-layout through a per-wave LDS tile (stride 18 floats).

#define GRU_I   8
#define GRU_H   16
#define GRU_T   512
#define GRU_B   8192
#define GRU_OUT 10

typedef float v2f __attribute__((ext_vector_type(2)));
typedef float v8f __attribute__((ext_vector_type(8)));

static __device__ __forceinline__ v8f wmma4(v2f a, v2f b, v8f c) {
    // D = A(16x4 f32) * B(4x16 f32) + C(16x16 f32)
    return __builtin_amdgcn_wmma_f32_16x16x4_f32(
        /*neg_a=*/false, a, /*neg_b=*/false, b,
        /*c_mod=*/(short)0, c, /*reuse_a=*/false, /*reuse_b=*/false);
}

static __device__ __forceinline__ float fast_sigmoid(float x) {
    return __builtin_amdgcn_rcpf(1.0f + __expf(-x));
}
static __device__ __forceinline__ float fast_tanh(float x) {
    // tanh(x) = 1 - 2/(exp(2x)+1)
    return 1.0f - 2.0f * __builtin_amdgcn_rcpf(__expf(2.0f * x) + 1.0f);
}

static __device__ __forceinline__ v8f splat8(float v) {
    v8f r;
    r[0] = v; r[1] = v; r[2] = v; r[3] = v;
    r[4] = v; r[5] = v; r[6] = v; r[7] = v;
    return r;
}

// LDS row stride (floats) for the per-wave 16x16 h tile: 18 keeps ds_load_b64
// 8-byte aligned (even) and spreads the 64 banks (18*m mod 64 all distinct).
#define LDS_STRIDE 18

__global__ __launch_bounds__(256) void gru_wmma_kernel(
    const float* __restrict__ x,     // [B, T, I]
    const float* __restrict__ W_ih,  // [3H, I]
    const float* __restrict__ W_hh,  // [3H, H]
    const float* __restrict__ b_ih,  // [3H]
    const float* __restrict__ b_hh,  // [3H]
    const float* __restrict__ W_fc,  // [OUT, H]
    const float* __restrict__ b_fc,  // [OUT]
    float* __restrict__ out)         // [B, OUT]
{
    __shared__ float lds_h[8 * 16 * LDS_STRIDE];

    const int lane   = threadIdx.x & 31;
    const int waveId = threadIdx.x >> 5;
    const int l15    = lane & 15;
    const int hi     = lane >> 4;           // 0: lanes 0-15, 1: lanes 16-31
    const int kofs   = hi ? 2 : 0;          // A/B layout: hi half holds K+2,K+3
    const int tile   = blockIdx.x * 8 + waveId;
    const int b0     = tile * 16;           // first batch row of this tile

    float* hlds = lds_h + waveId * (16 * LDS_STRIDE);

    // ---- B-matrix fragments (assumed layout mirrors A: VGPR j, lane l ->
    //      K = 4c + j + (hi?2:0), N = l&15) -----------------------------------
    // gx weights: B[k,n] = W_ih[g*16+n][k]
    v2f Bih[3][2];
#pragma unroll
    for (int g = 0; g < 3; ++g)
#pragma unroll
        for (int c = 0; c < 2; ++c)
            Bih[g][c] = *(const v2f*)(W_ih + (g * 16 + l15) * GRU_I + c * 4 + kofs);

    // gh weights: B[k,n] = W_hh[g*16+n][k]
    v2f Bhh[3][4];
#pragma unroll
    for (int g = 0; g < 3; ++g)
#pragma unroll
        for (int c = 0; c < 4; ++c)
            Bhh[g][c] = *(const v2f*)(W_hh + (g * 16 + l15) * GRU_H + c * 4 + kofs);

    // FC weights, N padded 10 -> 16 with zeros: B[k,n] = (n<10)? W_fc[n][k] : 0
    v2f Bfc[4];
#pragma unroll
    for (int c = 0; c < 4; ++c) {
        v2f w; w[0] = 0.0f; w[1] = 0.0f;
        if (l15 < GRU_OUT) w = *(const v2f*)(W_fc + l15 * GRU_H + c * 4 + kofs);
        Bfc[c] = w;
    }

    // Biases, broadcast over M (C layout: a lane's value depends only on N=l15)
    const float bias_r  = b_ih[l15]      + b_hh[l15];
    const float bias_z  = b_ih[16 + l15] + b_hh[16 + l15];
    const float bias_xn = b_ih[32 + l15];
    const float bias_hn = b_hh[32 + l15];
    const float bias_fc = (l15 < GRU_OUT) ? b_fc[l15] : 0.0f;

    // ---- h state: C/D layout (VGPR r, lanes<16 -> M=r, lanes>=16 -> M=r+8,
    //      N=l15).  Zero-init registers and the LDS tile. ---------------------
    v8f hC = splat8(0.0f);
#pragma unroll
    for (int r = 0; r < 8; ++r)
        hlds[(r + hi * 8) * LDS_STRIDE + l15] = 0.0f;
    __asm__ volatile("" ::: "memory");

    const float* xrow = x + ((size_t)(b0 + l15) * GRU_T) * GRU_I;

    for (int t = 0; t < GRU_T; ++t) {
        // x A-fragments: lane l -> M=l15, K = 4c + kofs + {0,1}
        v2f ax0 = *(const v2f*)(xrow + 0 + kofs);
        v2f ax1 = *(const v2f*)(xrow + 4 + kofs);
        __builtin_prefetch(xrow + 2 * GRU_I, 0, 1);   // global_prefetch_b8

        // h A-fragments from LDS (previous step's h, transposed layout)
        v2f ah[4];
#pragma unroll
        for (int c = 0; c < 4; ++c)
            ah[c] = *(const v2f*)(hlds + l15 * LDS_STRIDE + c * 4 + kofs);

        v8f acc_r  = splat8(bias_r);
        v8f acc_z  = splat8(bias_z);
        v8f acc_xn = splat8(bias_xn);
        v8f acc_hn = splat8(bias_hn);

        // gx contributions (K = 8 -> 2 chunks)
        acc_r  = wmma4(ax0, Bih[0][0], acc_r);
        acc_r  = wmma4(ax1, Bih[0][1], acc_r);
        acc_z  = wmma4(ax0, Bih[1][0], acc_z);
        acc_z  = wmma4(ax1, Bih[1][1], acc_z);
        acc_xn = wmma4(ax0, Bih[2][0], acc_xn);
        acc_xn = wmma4(ax1, Bih[2][1], acc_xn);

        // gh contributions (K = 16 -> 4 chunks)
#pragma unroll
        for (int c = 0; c < 4; ++c) {
            acc_r  = wmma4(ah[c], Bhh[0][c], acc_r);
            acc_z  = wmma4(ah[c], Bhh[1][c], acc_z);
            acc_hn = wmma4(ah[c], Bhh[2][c], acc_hn);
        }

        // element-wise GRU update (all operands share C/D layout)
#pragma unroll
        for (int j = 0; j < 8; ++j) {
            float rg = fast_sigmoid(acc_r[j]);
            float zg = fast_sigmoid(acc_z[j]);
            float ng = fast_tanh(acc_xn[j] + rg * acc_hn[j]);
            hC[j] = ng + zg * (hC[j] - ng);
        }

        // publish h for next step's A-fragments (per-wave LDS is in-order;
        // only a compiler barrier is needed, no s_barrier)
#pragma unroll
        for (int r = 0; r < 8; ++r)
            hlds[(r + hi * 8) * LDS_STRIDE + l15] = hC[r];

        __asm__ volatile("" ::: "memory");
        xrow += GRU_I;
    }

    // ---- final FC: out = sigmoid(h @ W_fc^T + b_fc), N padded to 16 ---------
    v2f ahf[4];
#pragma unroll
    for (int c = 0; c < 4; ++c)
        ahf[c] = *(const v2f*)(hlds + l15 * LDS_STRIDE + c * 4 + kofs);

    v8f acc = splat8(bias_fc);
#pragma unroll
    for (int c = 0; c < 4; ++c)
        acc = wmma4(ahf[c], Bfc[c], acc);

    if (l15 < GRU_OUT) {
#pragma unroll
        for (int r = 0; r < 8; ++r) {
            int m = r + hi * 8;
            out[(size_t)(b0 + m) * GRU_OUT + l15] = fast_sigmoid(acc[r]);
        }
    }
}

extern "C" void kernel_launch(void* const* d_in, const int* in_sizes, int n_in,
                              void* d_out, int out_size, void* d_ws, size_t ws_size,
                              hipStream_t stream) {
    (void)in_sizes; (void)n_in; (void)d_ws; (void)ws_size; (void)out_size;
    const float* x    = (const float*)d_in[0];
    const float* W_ih = (const float*)d_in[1];
    const float* W_hh = (const float*)d_in[2];
    const float* b_ih = (const float*)d_in[3];
    const float* b_hh = (const float*)d_in[4];
    const float* W_fc = (const float*)d_in[5];
    const float* b_fc = (const float*)d_in[6];
    float* out = (float*)d_out;

    // 8192 batch / 16 per wave = 512 waves; 8 waves (256 threads) per block.
    dim3 grid(GRU_B / (16 * 8));
    dim3 block(256);
    gru_wmma_kernel<<<grid, block, 0, stream>>>(x, W_ih, W_hh, b_ih, b_hh,
                                                W_fc, b_fc, out);
}